// Spherical_CNN_Regression_26388279066906
// MI455X (gfx1250) — compile-verified
//
#include <hip/hip_runtime.h>
#include <hip/hip_bf16.h>
#include <math.h>

// ---------------------------------------------------------------------------
// Spherical CNN regression, fully fused for gfx1250 (MI455X).
// One workgroup (256 thr = 8 waves, wave32) handles 16 batch elements.
// Hidden state lives in LDS (64KB); every GEMM uses V_WMMA_F32_16X16X4_F32.
// Weights/parts are fetched through explicit global address-space pointers
// (global_load_*, LOADcnt only) to keep the LDS pipe free for CG-term loads.
// All vector data uses clang ext_vector types (builtin, addrspace-agnostic).
// ---------------------------------------------------------------------------

typedef __attribute__((ext_vector_type(2))) float v2f;
typedef __attribute__((ext_vector_type(4))) float v4f;
typedef __attribute__((ext_vector_type(8))) float v8f;

// explicit global-address-space pointers -> global_load instead of flat_load
typedef const __attribute__((address_space(1))) v2f* gf2p;
typedef const __attribute__((address_space(1))) float* gfp;

__device__ __forceinline__ v8f wmma4(v2f a, v2f b, v8f c) {
  // D = A(16x4 f32) * B(4x16 f32) + C(16x16 f32)
  return __builtin_amdgcn_wmma_f32_16x16x4_f32(false, a, false, b, (short)0, c,
                                               false, false);
}

__device__ __forceinline__ v8f zero8() {
  v8f z;
  for (int i = 0; i < 8; i++) z[i] = 0.0f;
  return z;
}

__device__ __forceinline__ v2f mk2(float a, float b) {
  v2f r;
  r.x = a;
  r.y = b;
  return r;
}

// ------------------------- compile-time CG tables --------------------------
constexpr double FACT[11] = {1., 1., 2., 6., 24., 120., 720., 5040., 40320.,
                             362880., 3628800.};

constexpr double csqrt_(double x) {
  double g = x > 1.0 ? x : 1.0;
  for (int i = 0; i < 32; i++) g = 0.5 * (g + x / g);
  return g;
}

constexpr double cg_coeff(int l1, int l2, int L, int m1, int m2) {
  int M = m1 + m2;
  if (M < -L || M > L) return 0.0;
  double pre = csqrt_((2 * L + 1) * FACT[l1 + l2 - L] * FACT[l1 - l2 + L] *
                      FACT[-l1 + l2 + L] / FACT[l1 + l2 + L + 1]);
  pre = pre * csqrt_(FACT[L + M] * FACT[L - M] * FACT[l1 - m1] *
                     FACT[l1 + m1] * FACT[l2 - m2] * FACT[l2 + m2]);
  double s = 0.0;
  for (int k = 0; k <= l1 + l2 - L; k++) {
    int t1 = l1 + l2 - L - k, t2 = l1 - m1 - k, t3 = l2 + m2 - k;
    int t4 = L - l2 + m1 + k, t5 = L - l1 - m2 + k;
    if (t1 < 0 || t2 < 0 || t3 < 0 || t4 < 0 || t5 < 0) continue;
    double d = FACT[k] * FACT[t1] * FACT[t2] * FACT[t3] * FACT[t4] * FACT[t5];
    s += ((k & 1) ? -1.0 : 1.0) / d;
  }
  return pre * s;
}

struct CGTables {
  int nblk[4];
  int bl1[4][11], bl2[4][11];
  int nterm[4][7][11];
  signed char ti[4][7][11][7], tj[4][7][11][7];
  float cg[4][7][11][7];
};

constexpr CGTables make_tables() {
  CGTables t{};
  for (int L = 0; L < 4; L++) {
    int nb = 0;
    for (int l1 = 0; l1 < 4; l1++)
      for (int l2 = 0; l2 < 4; l2++) {
        int d = l1 - l2;
        if (d < 0) d = -d;
        if (!(d <= L && L <= l1 + l2)) continue;
        t.bl1[L][nb] = l1;
        t.bl2[L][nb] = l2;
        for (int M = 0; M < 2 * L + 1; M++) {
          int nt = 0;
          for (int i = 0; i <= 2 * l1; i++) {
            int j = M + l1 + l2 - L - i;  // m1+m2 == M-L
            if (j < 0 || j > 2 * l2) continue;
            double c = cg_coeff(l1, l2, L, i - l1, j - l2);
            if (c > 1e-12 || c < -1e-12) {
              t.ti[L][M][nb][nt] = (signed char)i;
              t.tj[L][M][nb][nt] = (signed char)j;
              t.cg[L][M][nb][nt] = (float)c;
              nt++;
            }
          }
          t.nterm[L][M][nb] = nt;
        }
        nb++;
      }
    t.nblk[L] = nb;
  }
  return t;
}

__device__ constexpr CGTables CGT = make_tables();
// row index (0..15) <-> (l, m) across the 1+3+5+7 rows
__device__ constexpr int ROW2L[16] = {0, 1, 1, 1, 2, 2, 2, 2, 2, 3, 3, 3, 3, 3, 3, 3};
__device__ constexpr int ROW2M[16] = {0, 0, 1, 2, 0, 1, 2, 3, 4, 0, 1, 2, 3, 4, 5, 6};
__device__ constexpr int LOFF[4] = {0, 1, 4, 9};

// normalize each l-block over its m rows, per (batch, channel)
__device__ __forceinline__ void normalize_lds(v2f (*hsh)[16][32], int tid) {
  for (int idx = tid; idx < 2048; idx += 256) {
    int l = idx >> 9, rem = idx & 511, b = rem >> 5, c = rem & 31;
    int ro = LOFF[l], nm = 2 * l + 1;
    float s = 0.0f;
    for (int mm = 0; mm < nm; mm++) {
      v2f v = hsh[ro + mm][b][c];
      s += v.x * v.x + v.y * v.y;
    }
    float inv = 1.0f / sqrtf(s);
    for (int mm = 0; mm < nm; mm++) {
      v2f v = hsh[ro + mm][b][c];
      hsh[ro + mm][b][c] = mk2(v.x * inv, v.y * inv);
    }
  }
}

__global__ __launch_bounds__(256) void sph_cnn_kernel(
    const float* __restrict__ p0, const float* __restrict__ p1,
    const float* __restrict__ p2, const float* __restrict__ p3,
    const float* __restrict__ wi0, const float* __restrict__ wi1,
    const float* __restrict__ wi2, const float* __restrict__ wi3,
    const float* __restrict__ wa0, const float* __restrict__ wa1,
    const float* __restrict__ wa2, const float* __restrict__ wa3,
    const float* __restrict__ wb0, const float* __restrict__ wb1,
    const float* __restrict__ wb2, const float* __restrict__ wb3,
    const float* __restrict__ W1p, const float* __restrict__ b1p,
    const float* __restrict__ W2p, const float* __restrict__ b2p,
    float* __restrict__ out) {
  __shared__ v2f hsh[16][16][32];  // [row(l,m)][batch][chan] complex, 64KB

  const int tid = threadIdx.x;
  const int lane = tid & 31;
  const int w = tid >> 5;     // wave id 0..7
  const int h16 = lane >> 4;  // half-wave (selects k pair)
  const int ln = lane & 15;   // lane-within-16 (selects M row / N col)
  const int bbase = blockIdx.x * 16;

  const gf2p P[4] = {(gf2p)p0, (gf2p)p1, (gf2p)p2, (gf2p)p3};
  const gf2p WI[4] = {(gf2p)wi0, (gf2p)wi1, (gf2p)wi2, (gf2p)wi3};
  const gf2p WH[2][4] = {{(gf2p)wa0, (gf2p)wa1, (gf2p)wa2, (gf2p)wa3},
                         {(gf2p)wb0, (gf2p)wb1, (gf2p)wb2, (gf2p)wb3}};
  const gfp W1 = (gfp)W1p, b1 = (gfp)b1p, W2 = (gfp)W2p, b2 = (gfp)b2p;

  // ---- Stage 1: hidden[l] = parts[l] @ W_in[l]  (complex, via f32 WMMA) ----
  // 32 tasks (16 rows x 2 n-tiles) over 8 waves; GEMM M=batch16, K=16, N=16
  for (int task = w; task < 32; task += 8) {
    const int row = task & 15, ntile = task >> 4;
    const int l = ROW2L[row], m = ROW2M[row], nlm = 2 * l + 1;
    const gf2p Pp = P[l];
    const gf2p Wl = WI[l];
    const int bg = bbase + ln;
    const int n = ntile * 16 + ln;
    v8f cre = zero8(), cim = zero8();
    for (int ks = 0; ks < 4; ks++) {
      const int t0 = 4 * ks + 2 * h16;
      v2f a0 = Pp[(bg * nlm + m) * 16 + t0];
      v2f a1 = Pp[(bg * nlm + m) * 16 + t0 + 1];
      v2f e0 = Wl[t0 * 32 + n];
      v2f e1 = Wl[(t0 + 1) * 32 + n];
      v2f ar = mk2(a0.x, a1.x), ai = mk2(a0.y, a1.y);
      v2f br = mk2(e0.x, e1.x), bi = mk2(e0.y, e1.y);
      cre = wmma4(ar, br, cre);
      cre = wmma4(-ai, bi, cre);
      cim = wmma4(ar, bi, cim);
      cim = wmma4(ai, br, cim);
    }
    for (int r = 0; r < 8; r++)
      hsh[row][r + 8 * h16][n] = mk2(cre[r], cim[r]);
  }
  __syncthreads();
  normalize_lds(hsh, tid);
  __syncthreads();

  // ---- Stage 2: two CG-product + channel-matmul layers ----
  for (int layer = 0; layer < 2; layer++) {
    v8f cre[2][2], cim[2][2];  // [row rr][ntile]
    for (int rr = 0; rr < 2; rr++) {
      const int row = w + 8 * rr;
      const int L = ROW2L[row], M = ROW2M[row];
      const gf2p Wm = WH[layer][L];
      cre[rr][0] = zero8(); cre[rr][1] = zero8();
      cim[rr][0] = zero8(); cim[rr][1] = zero8();
      const int nblk = CGT.nblk[L];
      for (int blk = 0; blk < nblk; blk++) {
        const int r1 = LOFF[CGT.bl1[L][blk]];
        const int r2 = LOFF[CGT.bl2[L][blk]];
        const int nt = CGT.nterm[L][M][blk];
        for (int ks = 0; ks < 8; ks++) {  // K = 32 channels
          const int k0 = 4 * ks + 2 * h16;
          // lane builds its own WMMA A-fragment of the CG product tile:
          // T[b][k] = sum_t cg * h_l1[b,i,k] * h_l2[b,j,k]   (complex)
          float tre0 = 0.f, tre1 = 0.f, tim0 = 0.f, tim1 = 0.f;
          for (int t = 0; t < nt; t++) {
            const int i = CGT.ti[L][M][blk][t];
            const int j = CGT.tj[L][M][blk][t];
            const float cg = CGT.cg[L][M][blk][t];
            // one b128 per side: {re(k0), im(k0), re(k0+1), im(k0+1)}
            const v4f x = *(const v4f*)&hsh[r1 + i][ln][k0];
            const v4f y = *(const v4f*)&hsh[r2 + j][ln][k0];
            tre0 += cg * (x.x * y.x - x.y * y.y);
            tim0 += cg * (x.x * y.y + x.y * y.x);
            tre1 += cg * (x.z * y.z - x.w * y.w);
            tim1 += cg * (x.z * y.w + x.w * y.z);
          }
          v2f ar = mk2(tre0, tre1), ai = mk2(tim0, tim1);
          const int kk = blk * 32 + k0;
          for (int ntile = 0; ntile < 2; ntile++) {
            const int n = ntile * 16 + ln;
            v2f e0 = Wm[kk * 32 + n];
            v2f e1 = Wm[(kk + 1) * 32 + n];
            v2f br = mk2(e0.x, e1.x), bi = mk2(e0.y, e1.y);
            cre[rr][ntile] = wmma4(ar, br, cre[rr][ntile]);
            cre[rr][ntile] = wmma4(-ai, bi, cre[rr][ntile]);
            cim[rr][ntile] = wmma4(ar, bi, cim[rr][ntile]);
            cim[rr][ntile] = wmma4(ai, br, cim[rr][ntile]);
          }
        }
      }
    }
    __syncthreads();  // all reads of old hidden complete
    for (int rr = 0; rr < 2; rr++) {
      const int row = w + 8 * rr;
      for (int ntile = 0; ntile < 2; ntile++) {
        const int n = ntile * 16 + ln;
        for (int r = 0; r < 8; r++)
          hsh[row][r + 8 * h16][n] = mk2(cre[rr][ntile][r], cim[rr][ntile][r]);
      }
    }
    __syncthreads();
    if (layer == 0) {
      normalize_lds(hsh, tid);
      __syncthreads();
    }
  }

  // ---- Stage 3/4: out = tanh(norms @ W1 + b1) @ W2 + b2 ----
  // GEMM M=batch16, K=128 (norms computed on the fly), N=256 (2 tiles/wave)
  float rsum[8];
  for (int r = 0; r < 8; r++) rsum[r] = 0.0f;
  for (int nt_i = 0; nt_i < 2; nt_i++) {
    const int ntile = w + 8 * nt_i;
    v8f acc = zero8();
    for (int ks = 0; ks < 32; ks++) {
      const int k0 = 4 * ks + 2 * h16;
      float av[2];
      for (int v = 0; v < 2; v++) {
        const int k = k0 + v;
        const int l = k >> 5, c = k & 31;
        const int ro = LOFF[l], nm = 2 * l + 1;
        float s = 0.0f;
        for (int mm = 0; mm < nm; mm++) {
          v2f t2 = hsh[ro + mm][ln][c];
          s += t2.x * t2.x + t2.y * t2.y;
        }
        av[v] = sqrtf(s);
      }
      v2f a = mk2(av[0], av[1]);
      const int n = ntile * 16 + ln;
      v2f bb = mk2(W1[k0 * 256 + n], W1[(k0 + 1) * 256 + n]);
      acc = wmma4(a, bb, acc);
    }
    const int n = ntile * 16 + ln;
    const float bias = b1[n], w2v = W2[n];
    for (int r = 0; r < 8; r++) rsum[r] += tanhf(acc[r] + bias) * w2v;
  }
  __syncthreads();  // hsh no longer needed; reuse as reduction scratch
  float* partial = (float*)&hsh[0][0][0];  // [8 waves][16 batch]
  for (int r = 0; r < 8; r++) {
    float v = rsum[r];
    v += __shfl_xor(v, 1);
    v += __shfl_xor(v, 2);
    v += __shfl_xor(v, 4);
    v += __shfl_xor(v, 8);  // deterministic reduce over 16 N-lanes
    if (ln == 0) partial[w * 16 + (8 * h16 + r)] = v;
  }
  __syncthreads();
  if (tid < 16) {
    float s = 0.0f;
    for (int ww = 0; ww < 8; ww++) s += partial[ww * 16 + tid];
    out[bbase + tid] = s + b2[0];
  }
}

extern "C" void kernel_launch(void* const* d_in, const int* in_sizes, int n_in,
                              void* d_out, int out_size, void* d_ws,
                              size_t ws_size, hipStream_t stream) {
  (void)n_in; (void)d_ws; (void)ws_size; (void)out_size;
  const int B = in_sizes[0] / 32;  // part_0 is [B,1,16,2]
  const int blocks = B / 16;
  sph_cnn_kernel<<<blocks, 256, 0, stream>>>(
      (const float*)d_in[0], (const float*)d_in[1], (const float*)d_in[2],
      (const float*)d_in[3], (const float*)d_in[4], (const float*)d_in[5],
      (const float*)d_in[6], (const float*)d_in[7], (const float*)d_in[8],
      (const float*)d_in[9], (const float*)d_in[10], (const float*)d_in[11],
      (const float*)d_in[12], (const float*)d_in[13], (const float*)d_in[14],
      (const float*)d_in[15], (const float*)d_in[16], (const float*)d_in[17],
      (const float*)d_in[18], (const float*)d_in[19], (float*)d_out);
}